// Model_16913581212004
// MI455X (gfx1250) — compile-verified
//
#include <hip/hip_runtime.h>

// ---------------------------------------------------------------------------
// CDNA5 (gfx1250) coref scorer. Strategy: algebraic factorization of both MLP
// first layers + f32 WMMA (V_WMMA_F32_16X16X4_F32) for all dense GEMMs.
// All GEMMs use N padded to 160 and zero-padded weights so tiles are guard-free.
// ---------------------------------------------------------------------------

typedef float v2f __attribute__((ext_vector_type(2)));
typedef float v8f __attribute__((ext_vector_type(8)));

#define T_TOK   2048
#define D_EMB   400
#define H2_DIM  400
#define MAXW    10
#define M_MENT  600
#define M_PAD   608          // 38 m-tiles of 16
#define A_ANTE  100
#define A_PAD   112          // 7 m-tiles of 16
#define FS      20
#define NPAD    160          // hidden 150 padded to 10 n-tiles of 16
#define KIN     1220         // mention embedding width (divisible by 4)

static __device__ __forceinline__ v8f wmma_f32(v2f a, v2f b, v8f c) {
  // D = A(16x4) * B(4x16) + C(16x16), all f32, wave32.
  return __builtin_amdgcn_wmma_f32_16x16x4_f32(
      /*neg_a=*/false, a, /*neg_b=*/false, b,
      /*c_mod=*/(short)0, c, /*reuse_a=*/false, /*reuse_b=*/false);
}

// ---------------------------------------------------------------------------
// head_logits[t] = dot(lstm[t], W_span) + b_span      (wave per token)
// ---------------------------------------------------------------------------
__global__ void head_logits_k(const float* __restrict__ lstm,
                              const float* __restrict__ wspan,
                              const float* __restrict__ bspan,
                              float* __restrict__ out) {
  int t = blockIdx.x;
  int lane = threadIdx.x;
  float acc = 0.f;
  for (int k = lane; k < H2_DIM; k += 32)
    acc += lstm[(size_t)t * H2_DIM + k] * wspan[k];
  for (int off = 16; off >= 1; off >>= 1) acc += __shfl_xor(acc, off, 32);
  if (lane == 0) out[t] = acc + bspan[0];
}

// ---------------------------------------------------------------------------
// Build ment_embs [M_PAD, 1220] = [start(400) | end(400) | width(20) | head(400)]
// Rows >= 600 zeroed (WMMA M-padding). Also records mention speaker id.
// ---------------------------------------------------------------------------
__global__ void build_ments_k(const float* __restrict__ doc,
                              const float* __restrict__ lstm,
                              const int* __restrict__ cstarts,
                              const int* __restrict__ cends,
                              const int* __restrict__ topidx,
                              const int* __restrict__ speaker_ids,
                              const float* __restrict__ wtable,
                              const float* __restrict__ head_logits,
                              float* __restrict__ ment_embs,
                              int* __restrict__ ment_spk) {
  int m = blockIdx.x;
  int t = threadIdx.x;
  float* row = ment_embs + (size_t)m * KIN;
  if (m >= M_MENT) {
    for (int i = t; i < KIN; i += blockDim.x) row[i] = 0.f;
    if (t == 0) ment_spk[m] = 0;
    return;
  }
  int ci = topidx[m];
  int s = cstarts[ci], e = cends[ci];
  int width = e - s + 1;
  for (int i = t; i < H2_DIM; i += blockDim.x) {
    row[i]          = lstm[(size_t)s * H2_DIM + i];
    row[H2_DIM + i] = lstm[(size_t)e * H2_DIM + i];
  }
  for (int i = t; i < FS; i += blockDim.x)
    row[2 * H2_DIM + i] = wtable[(width - 1) * FS + i];

  __shared__ float attn[MAXW];
  if (t == 0) {
    float lg[MAXW];
    float mx = -3.0e38f;
    for (int j = 0; j < MAXW; ++j) {
      int idx = s + j; if (idx > T_TOK - 1) idx = T_TOK - 1;
      lg[j] = (j < width) ? head_logits[idx] : -__builtin_inff();
      if (lg[j] > mx) mx = lg[j];
    }
    float se = 0.f;
    for (int j = 0; j < MAXW; ++j) { lg[j] = __expf(lg[j] - mx); se += lg[j]; }
    float inv = 1.f / se;
    for (int j = 0; j < MAXW; ++j) attn[j] = lg[j] * inv;
  }
  __syncthreads();
  for (int d = t; d < D_EMB; d += blockDim.x) {
    float acc = 0.f;
    for (int j = 0; j < MAXW; ++j) {
      int idx = s + j; if (idx > T_TOK - 1) idx = T_TOK - 1;
      acc += attn[j] * doc[(size_t)idx * D_EMB + d];
    }
    row[2 * H2_DIM + FS + d] = acc;
  }
  if (t == 0) ment_spk[m] = speaker_ids[s];
}

// ---------------------------------------------------------------------------
// Zero-pad weights: src rows [srcRowOff, srcRowOff+K) of a [*,150] row-major
// matrix -> dst [Kpad, 160], zeros elsewhere.
// ---------------------------------------------------------------------------
__global__ void pad_w_k(const float* __restrict__ src, int srcRowOff,
                        int K, int Kpad, float* __restrict__ dst) {
  int t = blockIdx.x * blockDim.x + threadIdx.x;
  if (t >= Kpad * NPAD) return;
  int r = t / NPAD, c = t % NPAD;
  dst[t] = (r < K && c < 150) ? src[(size_t)(srcRowOff + r) * 150 + c] : 0.f;
}

// ---------------------------------------------------------------------------
// Project feature tables through their as_w1 slices:
//  rows 0-1: speaker_proj[2,160], row 2: genre_vec[160], rows 3-12: dist_proj[10,160]
// ---------------------------------------------------------------------------
__global__ void feat_proj_k(const float* __restrict__ sptab,
                            const float* __restrict__ gentab,
                            const float* __restrict__ disttab,
                            const float* __restrict__ as_w1,
                            const int* __restrict__ genre_id,
                            float* __restrict__ sp_proj,
                            float* __restrict__ gen_vec,
                            float* __restrict__ dist_proj) {
  int t = blockIdx.x * blockDim.x + threadIdx.x;
  if (t >= 13 * NPAD) return;
  int rowsel = t / NPAD, n = t % NPAD;
  float acc = 0.f;
  if (n < 150) {
    const float* tab; int base;
    if (rowsel < 2)       { tab = sptab + rowsel * FS;        base = 3 * KIN; }
    else if (rowsel == 2) { tab = gentab + genre_id[0] * FS;  base = 3 * KIN + FS; }
    else                  { tab = disttab + (rowsel - 3) * FS; base = 3 * KIN + 2 * FS; }
    for (int f = 0; f < FS; ++f)
      acc += tab[f] * as_w1[(size_t)(base + f) * 150 + n];
  }
  if (rowsel < 2)       sp_proj[rowsel * NPAD + n] = acc;
  else if (rowsel == 2) gen_vec[n] = acc;
  else                  dist_proj[(rowsel - 3) * NPAD + n] = acc;
}

// ---------------------------------------------------------------------------
// Generic f32 WMMA GEMM: C[M,160] = op(A[M,K] @ B[K,160] + bias)
// One wave per 16x16 tile. grid = (ntiles, mtiles), block = 32.
// mode: 0 = none, 1 = +bias, 2 = relu(+bias). bias valid for n < biasN.
// ---------------------------------------------------------------------------
__global__ void gemm_wmma_k(const float* __restrict__ A, int lda,
                            const float* __restrict__ B,
                            float* __restrict__ C, int K,
                            const float* __restrict__ bias, int biasN,
                            int mode) {
  int lane = threadIdx.x & 31;
  int h = lane >> 4, r = lane & 15;
  int n0 = blockIdx.x * 16;
  int m0 = blockIdx.y * 16;
  int n = n0 + r;
  const float* Arow = A + (size_t)(m0 + r) * lda;
  v8f acc = {0.f, 0.f, 0.f, 0.f, 0.f, 0.f, 0.f, 0.f};
  for (int k = 0; k < K; k += 4) {
    int ka = k + 2 * h;
    v2f a, b;
    a.x = Arow[ka];
    a.y = Arow[ka + 1];
    b.x = B[(size_t)ka * NPAD + n];
    b.y = B[(size_t)(ka + 1) * NPAD + n];
    acc = wmma_f32(a, b, acc);
  }
  float bv = 0.f;
  if (mode != 0 && n < biasN) bv = bias[n];
  for (int j = 0; j < 8; ++j) {
    float v = acc[j] + bv;
    if (mode == 2) v = fmaxf(v, 0.f);
    C[(size_t)(m0 + j + 8 * h) * NPAD + n] = v;
  }
}

// ---------------------------------------------------------------------------
// ment_scores[m] = dot(h2m[m,0:150], ms_w3) + ms_b3   (wave per mention)
// ---------------------------------------------------------------------------
__global__ void ment_score_k(const float* __restrict__ h2,
                             const float* __restrict__ w3,
                             const float* __restrict__ b3,
                             float* __restrict__ out) {
  int m = blockIdx.x;
  int lane = threadIdx.x;
  float acc = 0.f;
  for (int k = lane; k < 150; k += 32)
    acc += h2[(size_t)m * NPAD + k] * w3[k];
  for (int off = 16; off >= 1; off >>= 1) acc += __shfl_xor(acc, off, 32);
  if (lane == 0) out[m] = acc + b3[0];
}

// ---------------------------------------------------------------------------
// Fused antecedent scorer. One block per mention m; 10 waves (n-tiles of 16).
// Phase 1: H1[a,n] = relu( (ment_m . ment_ant)@W1c + Pa[m]+Pb[ant]+feat )
//          - products staged cooperatively into LDS once per 16-ant tile.
// Phase 2: H2 = relu(H1 @ W2 + b2); score[a] += H2 . w3 via shuffle + ds_add.
// Final:   coref row assembly with antecedent mask.
// LDS: ment_m[1232] | stage[16*1224] | H1[112*160] | score[112]  (~152 KB)
// ---------------------------------------------------------------------------
#define SM_MENT   0
#define SM_STAGE  1232
#define STAGE_LD  1224
#define SM_H1     (SM_STAGE + 16 * STAGE_LD)
#define SM_SCORE  (SM_H1 + A_PAD * NPAD)
#define SM_TOTAL  (SM_SCORE + A_PAD)

__global__ void __launch_bounds__(320)
ante_fused_k(const float* __restrict__ ment_embs,
             const float* __restrict__ Pa, const float* __restrict__ Pb,
             const float* __restrict__ w1c, const float* __restrict__ w2a,
             const float* __restrict__ as_b2, const float* __restrict__ as_w3,
             const float* __restrict__ as_b3,
             const float* __restrict__ sp_proj, const float* __restrict__ gen_vec,
             const float* __restrict__ dist_proj,
             const int* __restrict__ antecedents, const int* __restrict__ ante_len,
             const int* __restrict__ ment_spk, const int* __restrict__ bins,
             const float* __restrict__ ment_scores,
             float* __restrict__ out) {
  extern __shared__ float smem[];
  __shared__ int s_ante[16];
  float* sm_ment  = smem + SM_MENT;
  float* sm_stage = smem + SM_STAGE;
  float* sm_h1    = smem + SM_H1;
  float* sm_score = smem + SM_SCORE;

  int m = blockIdx.x;
  int tid = threadIdx.x;
  int lane = tid & 31, wave = tid >> 5;       // 10 waves
  int h = lane >> 4, r = lane & 15;
  int n0 = wave * 16, n = n0 + r;

  for (int i = tid; i < KIN; i += 320)
    sm_ment[i] = ment_embs[(size_t)m * KIN + i];
  for (int i = tid; i < A_PAD; i += 320) sm_score[i] = 0.f;

  int my_spk = ment_spk[m];
  float genv = gen_vec[n];
  float pav = Pa[(size_t)m * NPAD + n];

  // ---- phase 1: similarity GEMM + feature fusion -> H1 in LDS ----
  for (int mt = 0; mt < 7; ++mt) {
    int m0 = mt * 16;
    __syncthreads();                        // previous tile's stage reads done
    if (tid < 16) {
      int a = m0 + tid; if (a > A_ANTE - 1) a = A_ANTE - 1;
      s_ante[tid] = antecedents[m * A_ANTE + a];
    }
    __syncthreads();
    for (int i = tid; i < 16 * KIN; i += 320) {
      int row = i / KIN, k = i - row * KIN;
      sm_stage[row * STAGE_LD + k] =
          ment_embs[(size_t)s_ante[row] * KIN + k] * sm_ment[k];
    }
    __syncthreads();

    v8f acc = {0.f, 0.f, 0.f, 0.f, 0.f, 0.f, 0.f, 0.f};
    for (int k = 0; k < KIN; k += 4) {
      int ka = k + 2 * h;
      v2f a, b;
      a.x = sm_stage[r * STAGE_LD + ka];
      a.y = sm_stage[r * STAGE_LD + ka + 1];
      b.x = w1c[(size_t)ka * NPAD + n];
      b.y = w1c[(size_t)(ka + 1) * NPAD + n];
      acc = wmma_f32(a, b, acc);
    }
    for (int j = 0; j < 8; ++j) {
      int rloc = j + 8 * h;                 // local row in this tile
      int ante = s_ante[rloc];
      int same = (ment_spk[ante] == my_spk) ? 1 : 0;
      int dist = m - ante;
      if (dist < 0) dist = 0;
      if (dist > 64) dist = 64;
      int bin = bins[dist];
      float v = acc[j] + pav + Pb[(size_t)ante * NPAD + n]
              + sp_proj[same * NPAD + n] + genv + dist_proj[bin * NPAD + n];
      sm_h1[(m0 + rloc) * NPAD + n] = fmaxf(v, 0.f);
    }
  }
  __syncthreads();

  // ---- phase 2: H2 = relu(H1 @ W2 + b2); score += H2 . w3 ----
  float b2v = (n < 150) ? as_b2[n] : 0.f;
  float w3v = (n < 150) ? as_w3[n] : 0.f;
  for (int mt = 0; mt < 7; ++mt) {
    int m0 = mt * 16;
    v8f acc = {0.f, 0.f, 0.f, 0.f, 0.f, 0.f, 0.f, 0.f};
    for (int k = 0; k < NPAD; k += 4) {
      int ka = k + 2 * h;
      v2f a, b;
      a.x = sm_h1[(m0 + r) * NPAD + ka];
      a.y = sm_h1[(m0 + r) * NPAD + ka + 1];
      b.x = w2a[(size_t)ka * NPAD + n];
      b.y = w2a[(size_t)(ka + 1) * NPAD + n];
      acc = wmma_f32(a, b, acc);
    }
    for (int j = 0; j < 8; ++j) {
      float part = fmaxf(acc[j] + b2v, 0.f) * w3v;
      for (int off = 8; off >= 1; off >>= 1) part += __shfl_xor(part, off, 32);
      if (r == 0) atomicAdd(&sm_score[m0 + j + 8 * h], part);
    }
  }
  __syncthreads();

  // ---- final: coref row assembly ----
  int len = ante_len[m];
  if (tid == 0) out[(size_t)m * (A_ANTE + 1)] = 0.f;
  if (tid >= 1 && tid <= A_ANTE) {
    int a = tid - 1;
    float v;
    if (a < len) {
      int ante = antecedents[m * A_ANTE + a];
      v = sm_score[a] + as_b3[0] + ment_scores[m] + ment_scores[ante];
    } else {
      v = -__builtin_inff();
    }
    out[(size_t)m * (A_ANTE + 1) + tid] = v;
  }
}

// ---------------------------------------------------------------------------
extern "C" void kernel_launch(void* const* d_in, const int* in_sizes, int n_in,
                              void* d_out, int out_size, void* d_ws, size_t ws_size,
                              hipStream_t stream) {
  const float* doc     = (const float*)d_in[0];
  const float* lstm    = (const float*)d_in[1];
  const int* cstarts   = (const int*)d_in[2];
  const int* cends     = (const int*)d_in[3];
  const int* topidx    = (const int*)d_in[4];
  const int* antecede  = (const int*)d_in[5];
  const int* antelen   = (const int*)d_in[6];
  const int* spk       = (const int*)d_in[7];
  const int* genre     = (const int*)d_in[8];
  const int* bins      = (const int*)d_in[9];
  const float* wtab    = (const float*)d_in[10];
  const float* sptab   = (const float*)d_in[11];
  const float* gentab  = (const float*)d_in[12];
  const float* disttab = (const float*)d_in[13];
  const float* Wspan   = (const float*)d_in[14];
  const float* bspan   = (const float*)d_in[15];
  const float* msw1    = (const float*)d_in[16];
  const float* msb1    = (const float*)d_in[17];
  const float* msw2    = (const float*)d_in[18];
  const float* msb2    = (const float*)d_in[19];
  const float* msw3    = (const float*)d_in[20];
  const float* msb3    = (const float*)d_in[21];
  const float* asw1    = (const float*)d_in[22];
  const float* asb1    = (const float*)d_in[23];
  const float* asw2    = (const float*)d_in[24];
  const float* asb2    = (const float*)d_in[25];
  const float* asw3    = (const float*)d_in[26];
  const float* asb3    = (const float*)d_in[27];
  float* out = (float*)d_out;

  float* ws = (float*)d_ws;
  size_t off = 0;
  auto alloc = [&](size_t nf) { float* p = ws + off; off += nf; return p; };
  float* head_log = alloc(T_TOK);
  float* ments    = alloc((size_t)M_PAD * KIN);
  float* h1m      = alloc((size_t)M_PAD * NPAD);
  float* h2m      = alloc((size_t)M_PAD * NPAD);
  float* Pa       = alloc((size_t)M_PAD * NPAD);
  float* Pb       = alloc((size_t)M_PAD * NPAD);
  float* w1m      = alloc((size_t)KIN * NPAD);
  float* w2m      = alloc((size_t)NPAD * NPAD);
  float* w1a      = alloc((size_t)KIN * NPAD);
  float* w1b      = alloc((size_t)KIN * NPAD);
  float* w1c      = alloc((size_t)KIN * NPAD);
  float* w2a      = alloc((size_t)NPAD * NPAD);
  float* spproj   = alloc(2 * NPAD);
  float* genvec   = alloc(NPAD);
  float* distproj = alloc(10 * NPAD);
  float* mscores  = alloc(M_PAD);
  int*   mspk     = (int*)alloc(M_PAD);

  // 1. head logits
  head_logits_k<<<T_TOK, 32, 0, stream>>>(lstm, Wspan, bspan, head_log);
  // 2. mention embeddings (padded to 608 rows)
  build_ments_k<<<M_PAD, 128, 0, stream>>>(doc, lstm, cstarts, cends, topidx,
                                           spk, wtab, head_log, ments, mspk);
  // 3. weight padding
  int padBig = (KIN * NPAD + 255) / 256;
  int padSml = (NPAD * NPAD + 255) / 256;
  pad_w_k<<<padBig, 256, 0, stream>>>(msw1, 0, KIN, KIN, w1m);
  pad_w_k<<<padSml, 256, 0, stream>>>(msw2, 0, 150, NPAD, w2m);
  pad_w_k<<<padBig, 256, 0, stream>>>(asw1, 0,        KIN, KIN, w1a);
  pad_w_k<<<padBig, 256, 0, stream>>>(asw1, KIN,      KIN, KIN, w1b);
  pad_w_k<<<padBig, 256, 0, stream>>>(asw1, 2 * KIN,  KIN, KIN, w1c);
  pad_w_k<<<padSml, 256, 0, stream>>>(asw2, 0, 150, NPAD, w2a);
  // 4. feature-table projections
  feat_proj_k<<<(13 * NPAD + 255) / 256, 256, 0, stream>>>(
      sptab, gentab, disttab, asw1, genre, spproj, genvec, distproj);
  // 5. WMMA GEMMs: mention scorer + antecedent precomputes
  dim3 gM(NPAD / 16, M_PAD / 16);
  gemm_wmma_k<<<gM, 32, 0, stream>>>(ments, KIN,  w1m, h1m, KIN,  msb1, 150, 2);
  gemm_wmma_k<<<gM, 32, 0, stream>>>(h1m,   NPAD, w2m, h2m, NPAD, msb2, 150, 2);
  gemm_wmma_k<<<gM, 32, 0, stream>>>(ments, KIN,  w1a, Pa,  KIN,  asb1, 150, 1);
  gemm_wmma_k<<<gM, 32, 0, stream>>>(ments, KIN,  w1b, Pb,  KIN,  nullptr, 0, 0);
  // 6. mention scores
  ment_score_k<<<M_MENT, 32, 0, stream>>>(h2m, msw3, msb3, mscores);
  // 7. fused antecedent scorer + output assembly
  ante_fused_k<<<M_MENT, 320, SM_TOTAL * sizeof(float), stream>>>(
      ments, Pa, Pb, w1c, w2a, asb2, asw3, asb3,
      spproj, genvec, distproj, antecede, antelen, mspk, bins, mscores, out);
}